// GCN_14302241096444
// MI455X (gfx1250) — compile-verified
//
#include <hip/hip_runtime.h>

#if defined(__HIP_DEVICE_COMPILE__) && !__has_builtin(__builtin_amdgcn_wmma_f32_16x16x4_f32)
#error "builtin __builtin_amdgcn_wmma_f32_16x16x4_f32 not available on device"
#endif

#define N_NODES 100000
#define N_EDGES 1600000
#define DH      128
#define DOUT    16
#define NGRAPH  512
#define K_DIM   128

typedef float v2f __attribute__((ext_vector_type(2)));
typedef float v8f __attribute__((ext_vector_type(8)));

// ---------------- small elementwise kernels ----------------

__global__ void gcn_fill(float* __restrict__ p, int n, float v) {
    int i = blockIdx.x * blockDim.x + threadIdx.x;
    if (i < n) p[i] = v;
}

__global__ void gcn_deg(const int* __restrict__ col, const float* __restrict__ ew,
                        float* __restrict__ deg, int E) {
    int e = blockIdx.x * blockDim.x + threadIdx.x;
    if (e < E) atomicAdd(&deg[col[e]], ew[e]);
}

__global__ void gcn_rsqrt(float* __restrict__ p, int n) {
    int i = blockIdx.x * blockDim.x + threadIdx.x;
    if (i < n) {
        float d = p[i];
        p[i] = (d > 0.0f) ? rsqrtf(d) : 0.0f;
    }
}

__global__ void gcn_init_bias(float* __restrict__ out, const float* __restrict__ b, int total) {
    int i = blockIdx.x * blockDim.x + threadIdx.x;
    if (i < total) out[i] = b[i & (DH - 1)];
}

__global__ void gcn_relu(float* __restrict__ p, int n) {
    int i = blockIdx.x * blockDim.x + threadIdx.x;
    if (i < n) p[i] = fmaxf(p[i], 0.0f);
}

// ---------------- WMMA fp32 GEMM: OUT[M, NT*16] = A[M,128] @ W[128, NT*16] (+bias) ----------------
// One wave -> one 16-row tile, all NT column tiles (A fragment reused).
// W staged in LDS (64KB f32 for NT=8).

template <int NT>
__global__ __launch_bounds__(256) void gcn_gemm_wmma(const float* __restrict__ A,
                                                     const float* __restrict__ W,
                                                     const float* __restrict__ bias,
                                                     float* __restrict__ OUT, int M) {
    constexpr int NC = NT * 16;
    __shared__ float sW[K_DIM * NC];

    const int tid = threadIdx.x;
    for (int i = tid; i < K_DIM * NC; i += 256) sW[i] = W[i];
    __syncthreads();

    const int lane = tid & 31;
    const int wave = tid >> 5;
    const int rowTile = blockIdx.x * 8 + wave;
    const int mtiles = M >> 4;
    if (rowTile >= mtiles) return;   // whole-wave exit; WMMA below runs with EXEC all-ones

    v8f acc[NT] = {};

    const int mrow = (rowTile << 4) + (lane & 15);
    const int koff = (lane >> 4) << 1;   // 0 for lanes 0-15, 2 for lanes 16-31
    const int ncol = lane & 15;
    const float* __restrict__ arow = A + (size_t)mrow * K_DIM;

    #pragma unroll 4
    for (int k = 0; k < K_DIM; k += 4) {
        // A-matrix 16x4 f32 layout: lane L, VGPR j = A[L%16][k + 2*(L/16) + j]
        const float2 a2 = *(const float2*)(arow + k + koff);
        v2f afrag;
        afrag.x = a2.x;
        afrag.y = a2.y;
        #pragma unroll
        for (int ct = 0; ct < NT; ++ct) {
            // B-matrix 4x16 f32 layout: lane L, VGPR j = B[2*(L/16) + j][L%16]
            v2f bfrag;
            bfrag.x = sW[(k + koff + 0) * NC + ct * 16 + ncol];
            bfrag.y = sW[(k + koff + 1) * NC + ct * 16 + ncol];
            acc[ct] = __builtin_amdgcn_wmma_f32_16x16x4_f32(
                false, afrag, false, bfrag, (short)0, acc[ct], false, false);
        }
    }

    // C/D layout: VGPR r, lane L -> D[M = r + 8*(L/16)][N = L%16]
    const int mb = (rowTile << 4) + ((lane >> 4) << 3);
    #pragma unroll
    for (int ct = 0; ct < NT; ++ct) {
        const float bv = bias ? bias[ct * 16 + ncol] : 0.0f;
        #pragma unroll
        for (int r = 0; r < 8; ++r) {
            OUT[(size_t)(mb + r) * NC + ct * 16 + ncol] = acc[ct][r] + bv;
        }
    }
}

// ---------------- edge scatter: out[col] += h[row] * (dinv[row]*w*dinv[col]) ----------------
// One wave per (edge or self-loop); lane handles a float4 (32 lanes * 4 = 128 feats).

__global__ void gcn_scatter(const float* __restrict__ H, const int* __restrict__ rowi,
                            const int* __restrict__ coli, const float* __restrict__ ew,
                            const float* __restrict__ dinv, float* __restrict__ out) {
    const int tid = blockIdx.x * blockDim.x + threadIdx.x;
    const int w = tid >> 5;
    const int lane = tid & 31;
    const int total = N_EDGES + N_NODES;
    if (w >= total) return;
    int r, c;
    float wt;
    if (w < N_EDGES) {
        r = rowi[w];
        c = coli[w];
        wt = ew[w];
    } else {                 // self loop, weight 1
        r = w - N_EDGES;
        c = r;
        wt = 1.0f;
    }
    const float nrm = dinv[r] * wt * dinv[c];
    const float4 h4 = ((const float4*)(H + (size_t)r * DH))[lane];
    float* o = out + (size_t)c * DH + lane * 4;
    atomicAdd(o + 0, h4.x * nrm);
    atomicAdd(o + 1, h4.y * nrm);
    atomicAdd(o + 2, h4.z * nrm);
    atomicAdd(o + 3, h4.w * nrm);
}

// ---------------- global mean pool ----------------

__global__ void gcn_pool(const float* __restrict__ H, const int* __restrict__ batch,
                         float* __restrict__ sums, float* __restrict__ cnt) {
    const int tid = blockIdx.x * blockDim.x + threadIdx.x;
    const int w = tid >> 5;
    const int lane = tid & 31;
    if (w >= N_NODES) return;
    const int g = batch[w];
    if (lane == 0) atomicAdd(&cnt[g], 1.0f);
    const float4 h4 = ((const float4*)(H + (size_t)w * DH))[lane];
    float* s = sums + (size_t)g * DH + lane * 4;
    atomicAdd(s + 0, h4.x);
    atomicAdd(s + 1, h4.y);
    atomicAdd(s + 2, h4.z);
    atomicAdd(s + 3, h4.w);
}

__global__ void gcn_pool_div(float* __restrict__ sums, const float* __restrict__ cnt) {
    const int i = blockIdx.x * blockDim.x + threadIdx.x;
    if (i < NGRAPH * DH) {
        const float c = cnt[i >> 7];
        sums[i] = sums[i] / fmaxf(c, 1.0f);
    }
}

// ---------------- launch ----------------

extern "C" void kernel_launch(void* const* d_in, const int* in_sizes, int n_in,
                              void* d_out, int out_size, void* d_ws, size_t ws_size,
                              hipStream_t stream) {
    (void)in_sizes; (void)n_in; (void)out_size; (void)ws_size;

    const float* x     = (const float*)d_in[0];
    const int*   ei    = (const int*)d_in[1];
    const float* ew    = (const float*)d_in[2];
    const int*   batch = (const int*)d_in[3];
    const float* W1    = (const float*)d_in[4];
    const float* b1    = (const float*)d_in[5];
    const float* W2    = (const float*)d_in[6];
    const float* b2    = (const float*)d_in[7];
    const float* W3    = (const float*)d_in[8];
    const float* b3    = (const float*)d_in[9];
    const float* Wlin  = (const float*)d_in[10];
    const float* blin  = (const float*)d_in[11];
    float* out = (float*)d_out;

    float* ws   = (float*)d_ws;
    float* dinv = ws;                                   // N
    float* T    = dinv + N_NODES;                       // N*DH (GEMM result)
    float* S    = T + (size_t)N_NODES * DH;             // N*DH (scatter accumulator)
    float* sums = S + (size_t)N_NODES * DH;             // G*DH
    float* cnt  = sums + (size_t)NGRAPH * DH;           // G

    const int* rowp = ei;
    const int* colp = ei + N_EDGES;

    const int nthr = 256;
    const int NE_W = (N_EDGES + N_NODES + 7) / 8;       // scatter blocks (8 waves/block)
    const int ND_B = (N_NODES * DH + nthr - 1) / nthr;
    const int gemmBlocks = ((N_NODES >> 4) + 7) / 8;    // 6250 row tiles / 8 waves

    // --- gcn_norm: deg (with self loop), then dinv = rsqrt(deg) ---
    gcn_fill<<<(N_NODES + nthr - 1) / nthr, nthr, 0, stream>>>(dinv, N_NODES, 1.0f);
    gcn_deg<<<(N_EDGES + nthr - 1) / nthr, nthr, 0, stream>>>(colp, ew, dinv, N_EDGES);
    gcn_rsqrt<<<(N_NODES + nthr - 1) / nthr, nthr, 0, stream>>>(dinv, N_NODES);

    // --- layer 1: T = x@W1; S = bias + scatter(T); relu ---
    gcn_gemm_wmma<8><<<gemmBlocks, nthr, 0, stream>>>(x, W1, nullptr, T, N_NODES);
    gcn_init_bias<<<ND_B, nthr, 0, stream>>>(S, b1, N_NODES * DH);
    gcn_scatter<<<NE_W, nthr, 0, stream>>>(T, rowp, colp, ew, dinv, S);
    gcn_relu<<<ND_B, nthr, 0, stream>>>(S, N_NODES * DH);

    // --- layer 2 ---
    gcn_gemm_wmma<8><<<gemmBlocks, nthr, 0, stream>>>(S, W2, nullptr, T, N_NODES);
    gcn_init_bias<<<ND_B, nthr, 0, stream>>>(S, b2, N_NODES * DH);
    gcn_scatter<<<NE_W, nthr, 0, stream>>>(T, rowp, colp, ew, dinv, S);
    gcn_relu<<<ND_B, nthr, 0, stream>>>(S, N_NODES * DH);

    // --- layer 3 (no relu) ---
    gcn_gemm_wmma<8><<<gemmBlocks, nthr, 0, stream>>>(S, W3, nullptr, T, N_NODES);
    gcn_init_bias<<<ND_B, nthr, 0, stream>>>(S, b3, N_NODES * DH);
    gcn_scatter<<<NE_W, nthr, 0, stream>>>(T, rowp, colp, ew, dinv, S);

    // --- global mean pool ---
    (void)hipMemsetAsync(sums, 0, (size_t)(NGRAPH * DH + NGRAPH) * sizeof(float), stream);
    gcn_pool<<<(N_NODES + 7) / 8, nthr, 0, stream>>>(S, batch, sums, cnt);
    gcn_pool_div<<<(NGRAPH * DH + nthr - 1) / nthr, nthr, 0, stream>>>(sums, cnt);

    // --- final linear: out[512,16] = hG @ Wlin + blin ---
    gcn_gemm_wmma<1><<<(NGRAPH >> 4) / 8, nthr, 0, stream>>>(sums, Wlin, blin, out, NGRAPH);
}